// Focal_SmoothL1_Loss_OD_21114059227558
// MI455X (gfx1250) — compile-verified
//
#include <hip/hip_runtime.h>
#include <hip/hip_bf16.h>

// Problem constants (match the reference).
constexpr int Bn   = 8;
constexpr int An   = 49104;
constexpr int Cn   = 80;
constexpr int Mn   = 64;
constexpr int TPB  = 256;                 // 8 wave32s per block
constexpr int ABLK = 256;                 // anchors per block
constexpr int NBLK = (An + ABLK - 1) / ABLK;  // 192

typedef float v2f __attribute__((ext_vector_type(2)));
typedef float v4f __attribute__((ext_vector_type(4)));
typedef float v8f __attribute__((ext_vector_type(8)));
typedef int   gv4i __attribute__((vector_size(16)));   // matches builtin's V4i

#if defined(__has_builtin)
#if __has_builtin(__builtin_amdgcn_wmma_f32_16x16x4_f32)
#define USE_WMMA_RED 1
#endif
#if __has_builtin(__builtin_amdgcn_global_load_async_to_lds_b128)
#define HAVE_ASYNC_LDS 1
#endif
#endif

// Wave32-wide sum using V_WMMA_F32_16X16X4_F32.
// A layout (32-bit A 16x4): lanes 0-15 -> (M=lane, K=0 in VGPR0, K=1 in VGPR1),
// lanes 16-31 -> (M=lane-16, K=2 in VGPR0, K=3 in VGPR1).
// With A.vgpr0 = v, A.vgpr1 = 0 and B = all-ones:
//   D[m,n] = v[m] + v[m+16]  (independent of n).
// Each lane's 8 D-VGPRs cover M = {0..7} (lanes 0-15) or {8..15} (lanes 16-31),
// so sum(D VGPRs) + shfl_xor(16) = full 32-lane sum in every lane.
// EXEC must be all ones: called unconditionally by fully-active blocks.
__device__ __forceinline__ float wave_sum32(float v) {
#ifdef USE_WMMA_RED
    v2f a;  a[0] = v;    a[1] = 0.0f;
    v2f bm; bm[0] = 1.0f; bm[1] = 1.0f;
    v8f c = {};
    v8f d = __builtin_amdgcn_wmma_f32_16x16x4_f32(
        false, a, false, bm, (short)0, c, false, false);
    float s = ((d[0] + d[1]) + (d[2] + d[3])) + ((d[4] + d[5]) + (d[6] + d[7]));
    s += __shfl_xor(s, 16, 32);
    return s;
#else
    for (int off = 16; off > 0; off >>= 1) v += __shfl_xor(v, off, 32);
    return v;
#endif
}

#ifdef HAVE_ASYNC_LDS
__device__ __forceinline__ void wait_async0() {
#if __has_builtin(__builtin_amdgcn_s_wait_asynccnt)
    __builtin_amdgcn_s_wait_asynccnt(0);
#else
    asm volatile("s_wait_asynccnt 0" ::: "memory");
#endif
}
#endif

__global__ __launch_bounds__(TPB) void od_loss_main(
    const float* __restrict__ reg,   // [B, A, 4]
    const float* __restrict__ cls,   // [B, A, C]
    const float* __restrict__ anc,   // [1, A, 4]
    const float* __restrict__ gt,    // [B, M, 5]
    float* __restrict__ part)        // [B*NBLK][3] = {cls_sum, reg_sum, npos}
{
    __shared__ __align__(16) float s_gt[Mn * 5];
    __shared__ int   s_info[ABLK];   // bits[1:0]: 0=ignore 1=neg 2=pos ; bits[9:2]: class id
    __shared__ float s_w[TPB / 32][3];

    const int tid = threadIdx.x;
    const int b   = blockIdx.y;
    const int a0  = blockIdx.x * ABLK;

    // Stage GT boxes for this image into LDS (tiny: 1280 B).
#ifdef HAVE_ASYNC_LDS
    // CDNA5 async copy: 80 lanes x 16 B, ASYNCcnt-tracked, no VGPR round-trip.
    if (tid < (Mn * 5) / 4) {
        const float* gsrc = gt + (size_t)b * Mn * 5 + tid * 4;
        __builtin_amdgcn_global_load_async_to_lds_b128(
            (__attribute__((address_space(1))) gv4i*)gsrc,
            (__attribute__((address_space(3))) gv4i*)&s_gt[tid * 4],
            0, 0);
    }
    wait_async0();
#else
    for (int i = tid; i < Mn * 5; i += TPB) s_gt[i] = gt[b * Mn * 5 + i];
#endif
    __syncthreads();

    // ---------------- Phase 1: per-anchor IoU argmax + regression loss ----------------
    const int  a      = a0 + tid;
    const bool active = (a < An);
    const int  aa     = active ? a : (An - 1);

    v4f ab = *(const v4f*)(anc + (size_t)aa * 4);
    const float ax1 = ab[0], ay1 = ab[1], ax2 = ab[2], ay2 = ab[3];
    const float aw  = fabsf(ax1 - ax2);
    const float ah  = fabsf(ay1 - ay2);
    const float acx = ax1 + 0.5f * aw;
    const float acy = ay1 + 0.5f * ah;
    const float aarea = (ax2 - ax1) * (ay2 - ay1);

    float best = -1e30f;
    int   bidx = 0;
#pragma unroll 4
    for (int m = 0; m < Mn; ++m) {
        const float gx1 = s_gt[m * 5 + 0], gy1 = s_gt[m * 5 + 1];
        const float gx2 = s_gt[m * 5 + 2], gy2 = s_gt[m * 5 + 3];
        const float gc  = s_gt[m * 5 + 4];
        const float iw = fmaxf(fminf(ax2, gx2) - fmaxf(ax1, gx1), 0.0f);
        const float ih = fmaxf(fminf(ay2, gy2) - fmaxf(ay1, gy1), 0.0f);
        const float inter = iw * ih;
        const float garea = (gx2 - gx1) * (gy2 - gy1);
        float iou = inter / (aarea + garea - inter);
        iou = (gc >= 0.0f) ? iou : -1.0f;
        if (iou > best) { best = iou; bidx = m; }   // strict '>' keeps first argmax
    }

    const bool pos = active && (best >= 0.5f);
    const bool neg = (best < 0.4f);
    const int  cid = (int)s_gt[bidx * 5 + 4];
    s_info[tid] = (!active ? 0 : (pos ? 2 : (neg ? 1 : 0))) | (cid << 2);

    // Regression targets from assigned GT (dx, dy, dh, dw channel order).
    const float gx1 = s_gt[bidx * 5 + 0], gy1 = s_gt[bidx * 5 + 1];
    const float gx2 = s_gt[bidx * 5 + 2], gy2 = s_gt[bidx * 5 + 3];
    const float gw0 = gx2 - gx1, gh0 = gy2 - gy1;
    const float gcx = gx1 + 0.5f * gw0, gcy = gy1 + 0.5f * gh0;
    const float gw = fmaxf(gw0, 1.0f), gh = fmaxf(gh0, 1.0f);
    const float t0 = (gcx - acx) / aw;          // tdx
    const float t1 = (gcy - acy) / ah;          // tdy
    const float t2 = __logf(gh / ah);           // tdh
    const float t3 = __logf(gw / aw);           // tdw

    v4f rg = *(const v4f*)(reg + ((size_t)b * An + aa) * 4);
    float sl1 = 0.0f;
    {
        const float t[4] = {t0, t1, t2, t3};
#pragma unroll
        for (int j = 0; j < 4; ++j) {
            const float d = fabsf(t[j] - rg[j]);
            sl1 += (d <= (1.0f / 9.0f)) ? (4.5f * d * d) : (d - 1.0f / 18.0f);
        }
    }
    float my_reg = pos ? sl1 : 0.0f;
    float my_np  = pos ? 1.0f : 0.0f;

    __syncthreads();   // s_info ready

    // ---------------- Phase 2: focal classification loss (coalesced stream) ----------------
    const int    acount = (An - a0 < ABLK) ? (An - a0) : ABLK;
    const int    total  = acount * Cn;                      // multiple of 4
    const float* cbase  = cls + ((size_t)b * An + a0) * Cn; // contiguous region
    float my_cls = 0.0f;

    for (int i = tid * 4; i < total; i += TPB * 4) {
        if (i + TPB * 4 < total)
            __builtin_prefetch(cbase + i + TPB * 4, 0, 0);  // global_prefetch_b8
        v4f p4 = __builtin_nontemporal_load((const v4f*)(cbase + i));
        const int al   = i / Cn;
        const int c    = i - al * Cn;
        const int info = s_info[al];
        const int md   = info & 3;
        if (md != 0) {
            const int cid2 = info >> 2;
#pragma unroll
            for (int j = 0; j < 4; ++j) {
                const float p = fminf(fmaxf(p4[j], 1e-4f), 1.0f - 1e-4f);
                const bool isone = (md == 2) && ((c + j) == cid2);
                const float x  = isone ? (1.0f - p) : p;
                const float af = isone ? 0.25f : 0.75f;
                const float lg = fmaxf(__logf(isone ? p : (1.0f - p)), -100.0f);
                my_cls += af * x * x * (-lg);
            }
        }
    }

    // ---------------- Phase 3: block reduction (WMMA per wave, fixed order) ----------------
    const float wc = wave_sum32(my_cls);
    const float wr = wave_sum32(my_reg);
    const float wn = wave_sum32(my_np);
    const int wave = tid >> 5, lane = tid & 31;
    if (lane == 0) { s_w[wave][0] = wc; s_w[wave][1] = wr; s_w[wave][2] = wn; }
    __syncthreads();
    if (tid == 0) {
        float c = 0.0f, r = 0.0f, n = 0.0f;
        for (int w = 0; w < TPB / 32; ++w) { c += s_w[w][0]; r += s_w[w][1]; n += s_w[w][2]; }
        const size_t blk = (size_t)b * gridDim.x + blockIdx.x;
        part[blk * 3 + 0] = c;
        part[blk * 3 + 1] = r;
        part[blk * 3 + 2] = n;
    }
}

// One wave32 per image: lanes stride the NBLK=192 block partials (6 iters),
// WMMA wave reduction, then thread 0 averages the 8 per-image losses.
__global__ __launch_bounds__(TPB) void od_loss_finalize(
    const float* __restrict__ part, float* __restrict__ out) {
    __shared__ float s_cls[Bn], s_reg[Bn];
    const int tid  = threadIdx.x;
    const int img  = tid >> 5;       // 8 waves == 8 images
    const int lane = tid & 31;

    float c = 0.0f, r = 0.0f, n = 0.0f;
    for (int k = lane; k < NBLK; k += 32) {   // exactly 6 iterations, no divergence
        const float* p = part + ((size_t)img * NBLK + k) * 3;
        c += p[0]; r += p[1]; n += p[2];
    }
    c = wave_sum32(c);
    r = wave_sum32(r);
    n = wave_sum32(n);

    if (lane == 0) {
        s_cls[img] = c / fmaxf(n, 1.0f);
        s_reg[img] = (n > 0.0f) ? (r / fmaxf(4.0f * n, 1.0f)) : 0.0f;
    }
    __syncthreads();
    if (tid == 0) {
        float cc = 0.0f, rr = 0.0f;
        for (int i = 0; i < Bn; ++i) { cc += s_cls[i]; rr += s_reg[i]; }
        out[0] = cc / (float)Bn;
        out[1] = (rr / (float)Bn) * 50.0f;
    }
}

extern "C" void kernel_launch(void* const* d_in, const int* in_sizes, int n_in,
                              void* d_out, int out_size, void* d_ws, size_t ws_size,
                              hipStream_t stream) {
    const float* reg = (const float*)d_in[0];   // [B, A, 4]
    const float* cls = (const float*)d_in[1];   // [B, A, C]
    const float* anc = (const float*)d_in[2];   // [1, A, 4]
    const float* gt  = (const float*)d_in[3];   // [B, M, 5]
    float* part = (float*)d_ws;                 // B*NBLK*3 floats = 18 KB

    dim3 grid(NBLK, Bn);
    od_loss_main<<<grid, TPB, 0, stream>>>(reg, cls, anc, gt, part);
    od_loss_finalize<<<1, TPB, 0, stream>>>(part, (float*)d_out);
}